// TagLoss_2_472446402690
// MI455X (gfx1250) — compile-verified
//
#include <hip/hip_runtime.h>

#define EPS  1e-4f
#define NKP  512      // keypoints per batch (N)

typedef float v2f __attribute__((ext_vector_type(2)));
typedef float v8f __attribute__((ext_vector_type(8)));

// wave32 butterfly reduction
__device__ __forceinline__ float waveReduce(float v) {
#pragma unroll
    for (int off = 16; off > 0; off >>= 1)
        v += __shfl_xor(v, off, 32);
    return v;
}

__global__ __launch_bounds__(NKP) void tagloss_batch_kernel(
    const float* __restrict__ tag1, const float* __restrict__ tag2,
    const int*   __restrict__ ind1, const int*  __restrict__ ind2,
    const int*   __restrict__ mask, float* __restrict__ ws, long hw)
{
    __shared__ __align__(16) float u[NKP];   // means (sentinel for masked)
    __shared__ float redN[16], redP[16], redS[16];

    const int b    = blockIdx.x;
    const int n    = threadIdx.x;
    const int lane = n & 31;
    const int wid  = n >> 5;

    // ---- gather phase ----
    const int i1 = ind1[b * NKP + n];
    const int i2 = ind2[b * NKP + n];
    const int mk = mask[b * NKP + n];

    const float t0   = tag1[(size_t)b * hw + i1];
    const float t1   = tag2[(size_t)b * hw + i2];
    const float mean = 0.5f * (t0 + t1);

    // Sentinel: masked entries get huge *distinct* values so every cross
    // pair involving them gives relu(1 - |diff|) == 0. Only the masked
    // diagonal contributes 1 each; corrected by (N - num) below.
    const float ui = mk ? mean : 1.0e8f * (float)(n + 1);
    u[n] = ui;

    // per-thread num / pull partials -> wave reduce -> LDS
    float numv  = waveReduce((float)mk);
    float pullv = waveReduce(mk ? 0.5f * (t0 - t1) * (t0 - t1) : 0.0f);
    if (lane == 0) { redN[wid] = numv; redP[wid] = pullv; }
    __syncthreads();   // publishes u[], redN, redP

    // ---- O(N^2) pairwise phase: thread n owns row i = n ----
    // Split the j-sweep in half: acc0/acc1 become this lane's two B-matrix
    // slots for the WMMA reduction. ds_load_b128 broadcast reads.
    float acc0 = 0.0f, acc1 = 0.0f;
    const float4* u4 = reinterpret_cast<const float4*>(u);
#pragma unroll 4
    for (int j4 = 0; j4 < NKP / 8; ++j4) {
        float4 v = u4[j4];
        acc0 += fmaxf(1.0f - fabsf(v.x - ui), 0.0f);
        acc0 += fmaxf(1.0f - fabsf(v.y - ui), 0.0f);
        acc0 += fmaxf(1.0f - fabsf(v.z - ui), 0.0f);
        acc0 += fmaxf(1.0f - fabsf(v.w - ui), 0.0f);
    }
#pragma unroll 4
    for (int j4 = NKP / 8; j4 < NKP / 4; ++j4) {
        float4 v = u4[j4];
        acc1 += fmaxf(1.0f - fabsf(v.x - ui), 0.0f);
        acc1 += fmaxf(1.0f - fabsf(v.y - ui), 0.0f);
        acc1 += fmaxf(1.0f - fabsf(v.z - ui), 0.0f);
        acc1 += fmaxf(1.0f - fabsf(v.w - ui), 0.0f);
    }

    // ---- WMMA wave reduction of the 64 partials (2 per lane) ----
    // D = ones(16x4) x B(4x16): every row of D equals the column sums, so
    // sum over lanes of D vgpr0 (rows M=0 and M=8) = 2 * sum of all B slots.
    // Layout-independent: A is all-ones, and we only need the total of B.
    // EXEC is all-ones here (uniform flow, full 512-thread block).
    v2f a;  a.x = 1.0f; a.y = 1.0f;
    v2f bb; bb.x = acc0; bb.y = acc1;
    v8f c = {};
    v8f d = __builtin_amdgcn_wmma_f32_16x16x4_f32(
        /*neg_a=*/false, a, /*neg_b=*/false, bb,
        /*c_mod=*/(short)0, c, /*reuse_a=*/false, /*reuse_b=*/false);
    float s = waveReduce(d[0]) * 0.5f;
    if (lane == 0) redS[wid] = s;
    __syncthreads();

    // ---- per-batch finalize (thread 0) ----
    if (n == 0) {
        float num = 0.0f, pull = 0.0f, S = 0.0f;
#pragma unroll
        for (int i = 0; i < 16; ++i) { num += redN[i]; pull += redP[i]; S += redS[i]; }
        const float pull_b  = pull / (num + EPS);
        const float S_valid = S - ((float)NKP - num);           // drop masked diagonal
        const float push_b  = (S_valid - num * num / (num + EPS))
                            / ((num - 1.0f) * num + EPS);
        ws[2 * b]     = pull_b;
        ws[2 * b + 1] = push_b;
    }
}

// Deterministic final reduction of the 128 per-batch (pull, push) pairs.
__global__ __launch_bounds__(128) void tagloss_finalize_kernel(
    const float* __restrict__ ws, float* __restrict__ out, int nb)
{
    __shared__ float redA[32], redB[32];
    const int n = threadIdx.x, lane = n & 31, wid = n >> 5;
    float p = (n < nb) ? ws[2 * n]     : 0.0f;
    float q = (n < nb) ? ws[2 * n + 1] : 0.0f;
    p = waveReduce(p);
    q = waveReduce(q);
    if (lane == 0) { redA[wid] = p; redB[wid] = q; }
    __syncthreads();
    if (n == 0) {
        float pull = 0.0f, push = 0.0f;
        const int nw = (blockDim.x + 31) >> 5;
        for (int i = 0; i < nw; ++i) { pull += redA[i]; push += redB[i]; }
        out[0] = pull;   // tuple order: (pull, push)
        out[1] = push;
    }
}

extern "C" void kernel_launch(void* const* d_in, const int* in_sizes, int n_in,
                              void* d_out, int out_size, void* d_ws, size_t ws_size,
                              hipStream_t stream) {
    const float* tag1 = (const float*)d_in[0];
    const float* tag2 = (const float*)d_in[1];
    const int*   ind1 = (const int*)d_in[2];
    const int*   ind2 = (const int*)d_in[3];
    const int*   mask = (const int*)d_in[4];
    float* ws  = (float*)d_ws;
    float* out = (float*)d_out;

    const int  B  = in_sizes[2] / NKP;        // 128
    const long HW = (long)(in_sizes[0] / B);  // 65536

    tagloss_batch_kernel<<<B, NKP, 0, stream>>>(tag1, tag2, ind1, ind2, mask, ws, HW);
    tagloss_finalize_kernel<<<1, 128, 0, stream>>>(ws, out, B);
}